// MoELinear_7808250544919
// MI455X (gfx1250) — compile-verified
//
#include <hip/hip_runtime.h>

#define NE    3
#define IN_F  512
#define OUT_F 512
#define TM    32            // tokens per tile (2 x 16-row WMMA sub-tiles)
#define LDSS  (IN_F + 4)    // padded LDS row stride (floats)

typedef float v2f __attribute__((ext_vector_type(2)));
typedef float v4f __attribute__((ext_vector_type(4)));
typedef float v8f __attribute__((ext_vector_type(8)));

// ws int layout: [0..2] counts, [3..6] tileOff[0..3] (tileOff[3]=totalTiles),
//                [7..9] scatter cursors, [10..15] pad, [16..) perm array (n + NE*TM)

__global__ void moe_init(int* __restrict__ ws) {
    if (threadIdx.x < 16) ws[threadIdx.x] = 0;
}

__global__ void moe_count(const int* __restrict__ ids, int n, int permLen,
                          int* __restrict__ ws, int* __restrict__ perm) {
    int i = blockIdx.x * blockDim.x + threadIdx.x;
    if (i < permLen) perm[i] = -1;
    if (i < n) {
        int e = ids[i];
        if (e < 0 || e >= NE) e = 0;
        atomicAdd(&ws[e], 1);
    }
}

__global__ void moe_prefix(int* __restrict__ ws) {
    int off = 0;
    ws[3] = 0;
    for (int e = 0; e < NE; ++e) {
        off += (ws[e] + TM - 1) / TM;   // padded tile count per expert
        ws[4 + e] = off;
    }
}

__global__ void moe_scatter(const int* __restrict__ ids, int n,
                            int* __restrict__ ws, int* __restrict__ perm) {
    int i = blockIdx.x * blockDim.x + threadIdx.x;
    if (i >= n) return;
    int e = ids[i];
    if (e < 0 || e >= NE) e = 0;
    int pos = atomicAdd(&ws[7 + e], 1);
    perm[ws[3 + e] * TM + pos] = i;
}

__global__ __launch_bounds__(256) void moe_gemm(
    const float* __restrict__ x, const float* __restrict__ W,
    const float* __restrict__ bias, const int* __restrict__ perm,
    const int* __restrict__ meta, float* __restrict__ out)
{
    __shared__ float As[TM][LDSS];   // 32 x 516 x 4B = 66 KB (CDNA5 WGP has 320 KB)
    __shared__ int   toks[TM];

    const int totalTiles = meta[6];
    const int t = blockIdx.x;
    if (t >= totalTiles) return;
    const int e = (t >= meta[4] ? 1 : 0) + (t >= meta[5] ? 1 : 0);

    const int tid  = threadIdx.x;
    const int lane = tid & 31;
    const int wave = tid >> 5;     // 0..7, each owns 64 out-features for all 32 rows
    const int half = lane >> 4;    // lane-half selects K sub-offset (ISA A/B layout)
    const int l16  = lane & 15;

    if (tid < TM) toks[tid] = perm[t * TM + tid];
    __syncthreads();

    // Stage A tile (32 x 512 fp32) into LDS: 8 threads/row, 16 x float4 each.
    {
        const int r  = tid >> 3;           // 0..31
        const int ct = tid & 7;            // 0..7
        const int tok = toks[r];
        const v4f* src = (const v4f*)(x + (size_t)(tok < 0 ? 0 : tok) * IN_F);
        const float zmask = (tok < 0) ? 0.0f : 1.0f;
        #pragma unroll
        for (int i = 0; i < 16; ++i) {
            const int c4 = ct + i * 8;     // float4 index 0..127
            v4f v = src[c4] * zmask;
            float* dst = &As[r][c4 * 4];
            dst[0] = v.x; dst[1] = v.y; dst[2] = v.z; dst[3] = v.w;
        }
    }
    __syncthreads();

    const int colBase = wave * 64;
    // A fragment source: rows l16 (tile 0) and l16+16 (tile 1); K sub-offset = half*2.
    const float* ap0 = &As[l16     ][half * 2];
    const float* ap1 = &As[l16 + 16][half * 2];
    // B fragment: B[k][n] = W_e[n][k]; per-lane row pointer, K walks along the row.
    const float* wp0 = W + (size_t)(e * OUT_F + colBase +  0 + l16) * IN_F + half * 2;
    const float* wp1 = W + (size_t)(e * OUT_F + colBase + 16 + l16) * IN_F + half * 2;
    const float* wp2 = W + (size_t)(e * OUT_F + colBase + 32 + l16) * IN_F + half * 2;
    const float* wp3 = W + (size_t)(e * OUT_F + colBase + 48 + l16) * IN_F + half * 2;

    v8f acc[2][4] = {};

    #pragma unroll 4
    for (int k = 0; k < IN_F; k += 4) {
        v2f a0 = *(const v2f*)(ap0 + k);
        v2f a1 = *(const v2f*)(ap1 + k);
        v2f b0 = *(const v2f*)(wp0 + k);
        v2f b1 = *(const v2f*)(wp1 + k);
        v2f b2 = *(const v2f*)(wp2 + k);
        v2f b3 = *(const v2f*)(wp3 + k);
        // D = A(16x4) * B(4x16) + C ; 8 args: (neg_a, A, neg_b, B, c_mod, C, reuse_a, reuse_b)
        acc[0][0] = __builtin_amdgcn_wmma_f32_16x16x4_f32(false, a0, false, b0, (short)0, acc[0][0], false, false);
        acc[0][1] = __builtin_amdgcn_wmma_f32_16x16x4_f32(false, a0, false, b1, (short)0, acc[0][1], false, false);
        acc[0][2] = __builtin_amdgcn_wmma_f32_16x16x4_f32(false, a0, false, b2, (short)0, acc[0][2], false, false);
        acc[0][3] = __builtin_amdgcn_wmma_f32_16x16x4_f32(false, a0, false, b3, (short)0, acc[0][3], false, false);
        acc[1][0] = __builtin_amdgcn_wmma_f32_16x16x4_f32(false, a1, false, b0, (short)0, acc[1][0], false, false);
        acc[1][1] = __builtin_amdgcn_wmma_f32_16x16x4_f32(false, a1, false, b1, (short)0, acc[1][1], false, false);
        acc[1][2] = __builtin_amdgcn_wmma_f32_16x16x4_f32(false, a1, false, b2, (short)0, acc[1][2], false, false);
        acc[1][3] = __builtin_amdgcn_wmma_f32_16x16x4_f32(false, a1, false, b3, (short)0, acc[1][3], false, false);
    }

    // Epilogue: C/D layout — VGPR r: lanes 0-15 => (M=r, N=lane), lanes 16-31 => (M=r+8, N=lane-16)
    #pragma unroll
    for (int j = 0; j < 4; ++j) {
        const int n = colBase + 16 * j + l16;
        const float bv = bias[e * OUT_F + n];
        #pragma unroll
        for (int tt = 0; tt < 2; ++tt) {
            #pragma unroll
            for (int r = 0; r < 8; ++r) {
                const int m = tt * 16 + r + half * 8;
                const int tok = toks[m];
                if (tok >= 0) out[(size_t)tok * OUT_F + n] = acc[tt][j][r] + bv;
            }
        }
    }
}

extern "C" void kernel_launch(void* const* d_in, const int* in_sizes, int n_in,
                              void* d_out, int out_size, void* d_ws, size_t ws_size,
                              hipStream_t stream) {
    const float* x    = (const float*)d_in[0];
    const int*   ids  = (const int*)d_in[1];
    const float* W    = (const float*)d_in[2];
    const float* bias = (const float*)d_in[3];
    float*       out  = (float*)d_out;

    const int n = in_sizes[0] / IN_F;
    int* ws   = (int*)d_ws;
    int* perm = ws + 16;
    const int permLen = n + NE * TM;

    const int thr = 256;
    moe_init   <<<1, 32, 0, stream>>>(ws);
    moe_count  <<<(permLen + thr - 1) / thr, thr, 0, stream>>>(ids, n, permLen, ws, perm);
    moe_prefix <<<1, 1, 0, stream>>>(ws);
    moe_scatter<<<(n + thr - 1) / thr, thr, 0, stream>>>(ids, n, ws, perm);

    const int maxTiles = (n + TM - 1) / TM + NE;   // padded upper bound; kernel trims via ws
    moe_gemm   <<<maxTiles, thr, 0, stream>>>(x, W, bias, perm, ws, out);
}